// DynamicGraphGenerator_17609365914276
// MI455X (gfx1250) — compile-verified
//
#include <hip/hip_runtime.h>
#include <math.h>

#define Bsz 16
#define Tlen 12
#define Ndim 2048
#define Ktop 20
#define EMB 16

typedef __attribute__((ext_vector_type(2))) float v2f;
typedef __attribute__((ext_vector_type(8))) float v8f;

// ---------------- K1: state mean + tanh embedding -> ws ----------------
__global__ void k_emb(const float* __restrict__ x, const float* __restrict__ W,
                      const float* __restrict__ bias, float* __restrict__ emb) {
    int g = blockIdx.x * blockDim.x + threadIdx.x;   // over B*N
    if (g >= Bsz * Ndim) return;
    int b = g / Ndim, n = g % Ndim;
    float s = 0.f;
    #pragma unroll
    for (int t = 0; t < Tlen; ++t)
        s += x[((size_t)b * Tlen + t) * Ndim + n];
    s *= (1.0f / Tlen);
    float tmp[EMB];
    #pragma unroll
    for (int h = 0; h < EMB; ++h)
        tmp[h] = tanhf(s * W[h] + bias[h]);
    float4* op = (float4*)(emb + (size_t)g * EMB);
    op[0] = make_float4(tmp[0], tmp[1], tmp[2], tmp[3]);
    op[1] = make_float4(tmp[4], tmp[5], tmp[6], tmp[7]);
    op[2] = make_float4(tmp[8], tmp[9], tmp[10], tmp[11]);
    op[3] = make_float4(tmp[12], tmp[13], tmp[14], tmp[15]);
}

// ---------------- K1b: physical-graph row-sum reciprocal ----------------
__global__ void k_rowsum(const float* __restrict__ phys, float* __restrict__ inv) {
    __shared__ float red[256];
    int n = blockIdx.x;
    float s = 0.f;
    for (int i = threadIdx.x; i < Ndim; i += 256)
        s += phys[(size_t)n * Ndim + i];
    red[threadIdx.x] = s;
    __syncthreads();
    for (int off = 128; off > 0; off >>= 1) {
        if ((int)threadIdx.x < off) red[threadIdx.x] += red[threadIdx.x + off];
        __syncthreads();
    }
    if (threadIdx.x == 0) inv[n] = 1.0f / (red[0] + 1e-8f);
}

// ---------------- K2: dense base  out = a * phys_norm (broadcast over batch) ----
__global__ void k_base(const float* __restrict__ phys, const float* __restrict__ inv,
                       const float* __restrict__ alpha, float* __restrict__ out) {
    size_t q = (size_t)blockIdx.x * blockDim.x + threadIdx.x;   // over B*N*N/4
    const int M4 = Ndim / 4;
    int m4 = (int)(q % M4);
    size_t rest = q / M4;
    int n = (int)(rest % Ndim);
    float a = 1.0f / (1.0f + __expf(-alpha[0]));
    float sc = a * inv[n];
    float4 p = ((const float4*)phys)[(size_t)n * M4 + m4];
    ((float4*)out)[q] = make_float4(p.x * sc, p.y * sc, p.z * sc, p.w * sc);
}

// ---------------- K3: WMMA Gram strip + ReLU + top-20 + softmax scatter --------
__global__ __launch_bounds__(256) void k_wmma_topk(const float* __restrict__ emb,
                                                   const float* __restrict__ alpha,
                                                   float* __restrict__ out) {
    extern __shared__ float smem[];
    float* rowdata = smem;                       // 16 * Ndim floats (128 KB strip)
    float* topvS   = smem + 16 * Ndim;           // 16 * Ktop
    int*   topiS   = (int*)(smem + 16 * Ndim + 16 * Ktop);

    const int b        = blockIdx.y;
    const int rowblock = blockIdx.x;             // 0 .. N/16-1
    const int tid  = threadIdx.x;
    const int wave = tid >> 5;                   // 8 waves
    const int lane = tid & 31;
    const int hs   = lane >> 4;                  // half-wave select (K-pair group)
    const int lr   = lane & 15;

    const float* embB0 = emb + (size_t)b * (Ndim * EMB);

    // A-fragments (rows of this block) — V_WMMA_F32_16X16X4_F32 layout:
    // lane<16 holds K = {s*4+0, s*4+1}; lane>=16 holds K = {s*4+2, s*4+3}
    const float* embA = embB0 + (size_t)(rowblock * 16 + lr) * EMB;
    float a0[4], a1[4];
    #pragma unroll
    for (int s = 0; s < 4; ++s) {
        a0[s] = embA[s * 4 + hs * 2];
        a1[s] = embA[s * 4 + hs * 2 + 1];
    }

    // Each wave computes 16 column tiles: 16x16 output per tile, K=16 via 4 WMMAs
    for (int t = 0; t < 16; ++t) {
        int ct = wave * 16 + t;                  // column tile index 0..127
        const float* embBp = embB0 + (size_t)(ct * 16 + lr) * EMB;
        v8f c = {0.f, 0.f, 0.f, 0.f, 0.f, 0.f, 0.f, 0.f};
        #pragma unroll
        for (int s = 0; s < 4; ++s) {
            v2f af; af.x = a0[s]; af.y = a1[s];
            v2f bf; bf.x = embBp[s * 4 + hs * 2]; bf.y = embBp[s * 4 + hs * 2 + 1];
            c = __builtin_amdgcn_wmma_f32_16x16x4_f32(
                    false, af, false, bf, (short)0, c, false, false);
        }
        // C layout: VGPR v, lanes<16 -> M=v, lanes>=16 -> M=v+8, N=lr. ReLU -> LDS.
        #pragma unroll
        for (int v = 0; v < 8; ++v) {
            float val = c[v];
            val = val > 0.f ? val : 0.f;
            rowdata[(v + 8 * hs) * Ndim + ct * 16 + lr] = val;
        }
    }
    __syncthreads();

    // Top-20 extraction: 16 threads per row, iterative argmax with shuffle butterfly.
    const int r = tid >> 4;                      // row within block 0..15
    const int j = tid & 15;                      // scanner lane within row team
    float* rowp = rowdata + r * Ndim;

    for (int it = 0; it < Ktop; ++it) {
        float best = -1.0f;                      // ReLU'd values are >= 0
        int bidx = 0;
        for (int i = 0; i < Ndim / 16; ++i) {
            float v = rowp[i * 16 + j];          // stride-16: conflict-free banks
            if (v > best) { best = v; bidx = i * 16 + j; }
        }
        #pragma unroll
        for (int off = 1; off < 16; off <<= 1) { // stays inside 16-lane half-wave
            float ov = __shfl_xor(best, off, 32);
            int   oi = __shfl_xor(bidx, off, 32);
            if (ov > best || (ov == best && oi < bidx)) { best = ov; bidx = oi; }
        }
        if (j == 0) {
            topvS[r * Ktop + it] = best;
            topiS[r * Ktop + it] = bidx;
            rowp[bidx] = -INFINITY;              // exclude from next extraction
        }
        __syncthreads();
    }

    // Softmax over the 20 survivors, scatter (1-a)*w into the dense base (unique idx).
    if (j == 0) {
        float aa = 1.0f / (1.0f + __expf(-alpha[0]));
        float m = topvS[r * Ktop + 0];           // first extracted == row max
        float e[Ktop];
        float s = 0.f;
        #pragma unroll
        for (int it = 0; it < Ktop; ++it) {
            e[it] = __expf(topvS[r * Ktop + it] - m);
            s += e[it];
        }
        float sc = (1.0f - aa) / s;
        size_t base = ((size_t)b * Ndim + rowblock * 16 + r) * (size_t)Ndim;
        #pragma unroll
        for (int it = 0; it < Ktop; ++it)
            out[base + topiS[r * Ktop + it]] += sc * e[it];
    }
}

extern "C" void kernel_launch(void* const* d_in, const int* in_sizes, int n_in,
                              void* d_out, int out_size, void* d_ws, size_t ws_size,
                              hipStream_t stream) {
    const float* x      = (const float*)d_in[0];   // [B,T,N,1]
    const float* A_phys = (const float*)d_in[1];   // [N,N]
    const float* W      = (const float*)d_in[2];   // [16,1]
    const float* bias   = (const float*)d_in[3];   // [16]
    const float* alpha  = (const float*)d_in[4];   // scalar
    float* out = (float*)d_out;                    // [B,N,N]

    float* emb = (float*)d_ws;                     // B*N*16 floats
    float* inv = emb + (size_t)Bsz * Ndim * EMB;   // N floats

    // K1: embeddings
    k_emb<<<(Bsz * Ndim + 255) / 256, 256, 0, stream>>>(x, W, bias, emb);
    // K1b: row-sum reciprocals of physical graph
    k_rowsum<<<Ndim, 256, 0, stream>>>(A_phys, inv);
    // K2: dense base write (bandwidth floor)
    {
        size_t total4 = (size_t)Bsz * Ndim * Ndim / 4;
        k_base<<<(unsigned)(total4 / 256), 256, 0, stream>>>(A_phys, inv, alpha, out);
    }
    // K3: WMMA strip + top-k + softmax scatter (depends on K1 & K2; stream-ordered)
    {
        dim3 g3(Ndim / 16, Bsz);
        size_t shmem = (size_t)(16 * Ndim + 16 * Ktop) * sizeof(float)
                     + (size_t)(16 * Ktop) * sizeof(int);
        k_wmma_topk<<<g3, 256, shmem, stream>>>(emb, alpha, out);
    }
}